// Condition9RotL_24799141167111
// MI455X (gfx1250) — compile-verified
//
#include <hip/hip_runtime.h>

typedef float v2f __attribute__((ext_vector_type(2)));
typedef float v8f __attribute__((ext_vector_type(8)));

#define N_PTS   262144
#define FEAT    128
#define HID     256
#define M_TILE  64
#define TPB     128          // 4 waves of 32

#define SF_STRIDE (FEAT + 4) // 132 floats: pad keeps 16B align + kills bank conflicts
#define SH_STRIDE (HID + 4)  // 260 floats

// One wave computes a [64 x 32] output slab: 4 M-subtiles x 2 N-tiles.
// A[64xK] from LDS (row-major, lda floats), B[KxN] global weights (row-major,
// ldw floats). Each B-fragment is loaded once and reused by 4 M-subtiles;
// each A-fragment is reused by the N-pair. 8 independent WMMA chains for ILP.
__device__ __forceinline__ void wmma_slab_4m2n(
    const float* sA, int lda,
    const float* __restrict__ W, int ldw,
    const float* __restrict__ bias,
    float* sD, int ldd,
    int n0, int K, int lane, bool relu)
{
    v8f acc[4][2] = {};
    const int r  = lane & 15;          // A row within subtile / B column
    const int ko = (lane >> 4) << 1;   // half-wave K pair select (0 or 2)
    const float* ap = sA + r * lda + ko;
    const float* wp = W + n0 + r;
#pragma unroll 2
    for (int k = 0; k < K; k += 4) {
        const float* w0 = wp + (size_t)(k + ko) * ldw;
        v2f b0, b1;
        b0.x = w0[0];    b0.y = w0[ldw];
        b1.x = w0[16];   b1.y = w0[ldw + 16];
#pragma unroll
        for (int m = 0; m < 4; ++m) {
            v2f a;
            a.x = ap[m * 16 * lda + k];
            a.y = ap[m * 16 * lda + k + 1];
            acc[m][0] = __builtin_amdgcn_wmma_f32_16x16x4_f32(false, a, false, b0,
                                                              (short)0, acc[m][0], false, false);
            acc[m][1] = __builtin_amdgcn_wmma_f32_16x16x4_f32(false, a, false, b1,
                                                              (short)0, acc[m][1], false, false);
        }
    }
    // C/D layout: VGPR i -> row (i or 8+i by half-wave), col = lane&15
    const int rsel = (lane >> 4) << 3;
    const float bb0 = bias[n0 + r];
    const float bb1 = bias[n0 + 16 + r];
#pragma unroll
    for (int m = 0; m < 4; ++m) {
#pragma unroll
        for (int i = 0; i < 8; ++i) {
            const int row = m * 16 + rsel + i;
            float v0 = acc[m][0][i] + bb0;
            float v1 = acc[m][1][i] + bb1;
            if (relu) { v0 = fmaxf(v0, 0.f); v1 = fmaxf(v1, 0.f); }
            sD[row * ldd + n0 + r]      = v0;
            sD[row * ldd + n0 + 16 + r] = v1;
        }
    }
}

__global__ __launch_bounds__(TPB) void fused_rot_mlp_polar(
    const float* __restrict__ rotation,  // [N,3,3]
    const float* __restrict__ feature,   // [N,128]
    const float* __restrict__ W1, const float* __restrict__ b1,
    const float* __restrict__ W2, const float* __restrict__ b2,
    const float* __restrict__ W3, const float* __restrict__ b3,
    float* __restrict__ out,             // [N,9]
    float* __restrict__ logdet)          // [N]
{
    extern __shared__ float smem[];
    float* sF  = smem;                          // [64][132]
    float* sH1 = sF  + M_TILE * SF_STRIDE;      // [64][260]
    float* sH2 = sH1 + M_TILE * SH_STRIDE;      // [64][260]

    const int tid  = threadIdx.x;
    const int lane = tid & 31;
    const int wave = tid >> 5;
    const int rowBase = blockIdx.x * M_TILE;

    // ---- stage feature tile [64 x 128] into LDS (float4, coalesced) ----
    for (int i = tid; i < M_TILE * (FEAT / 4); i += TPB) {
        const int r  = i >> 5;                  // 32 float4 per row
        const int c4 = (i & 31) << 2;
        const float4 v = *(const float4*)(feature + (size_t)(rowBase + r) * FEAT + c4);
        float* d = sF + r * SF_STRIDE + c4;
        d[0] = v.x; d[1] = v.y; d[2] = v.z; d[3] = v.w;
    }
    __syncthreads();

    // ---- layer 1: H1 = relu(F @ W1 + b1) ----
    // wave w owns N-tile pairs {2w,2w+1} and {2w+8,2w+9} for ALL 64 rows.
#pragma unroll
    for (int np = 0; np < 2; ++np) {
        const int n0 = (2 * wave + 8 * np) * 16;
        wmma_slab_4m2n(sF, SF_STRIDE, W1, HID, b1, sH1, SH_STRIDE, n0, FEAT, lane, true);
    }
    __syncthreads();

    // ---- layer 2: H2 = relu(H1 @ W2 + b2) ----
#pragma unroll
    for (int np = 0; np < 2; ++np) {
        const int n0 = (2 * wave + 8 * np) * 16;
        wmma_slab_4m2n(sH1, SH_STRIDE, W2, HID, b2, sH2, SH_STRIDE, n0, HID, lane, false ? false : true);
    }
    __syncthreads();

    // ---- layer 3: out9 = H2 @ W3 + b3 (N=9 in one 16-wide tile, per-wave M) ----
    {
        const int m0 = wave * 16;
        v8f acc = {};
        const int r   = lane & 15;
        const int ko  = (lane >> 4) << 1;
        const int rc  = (r < 9) ? r : 8;        // clamped index, masked value:
        const float msk = (r < 9) ? 1.f : 0.f;  // select, no branch -> EXEC stays all-1s
        const float* ap = sH2 + (m0 + r) * SH_STRIDE + ko;
#pragma unroll 4
        for (int k = 0; k < HID; k += 4) {
            v2f a; a.x = ap[k]; a.y = ap[k + 1];
            v2f b;
            b.x = W3[(size_t)(k + ko)     * 9 + rc] * msk;
            b.y = W3[(size_t)(k + ko + 1) * 9 + rc] * msk;
            acc = __builtin_amdgcn_wmma_f32_16x16x4_f32(false, a, false, b,
                                                        (short)0, acc, false, false);
        }
        const float bb = b3[rc] * msk;
        const int rsel = (lane >> 4) << 3;
        if (r < 9) {
#pragma unroll
            for (int i = 0; i < 8; ++i)
                sF[(m0 + rsel + i) * 16 + r] = acc[i] + bb;  // reuse sF as out9 staging
        }
    }
    __syncthreads();

    // ---- epilogue: one thread per point: (mat+I)@rot, polar projection ----
    if (tid < M_TILE) {
        const int g = rowBase + tid;
        float m[9];
#pragma unroll
        for (int j = 0; j < 9; ++j) m[j] = sF[tid * 16 + j];
        m[0] += 1.f; m[4] += 1.f; m[8] += 1.f;

        float rr[9];
#pragma unroll
        for (int j = 0; j < 9; ++j) rr[j] = rotation[(size_t)g * 9 + j];

        float X[9];
#pragma unroll
        for (int a = 0; a < 3; ++a)
#pragma unroll
            for (int c = 0; c < 3; ++c)
                X[a * 3 + c] = m[a * 3] * rr[c] + m[a * 3 + 1] * rr[3 + c] + m[a * 3 + 2] * rr[6 + c];

        // scaled Newton polar iteration: X <- 0.5*(g*X + (1/g)*X^{-T})
        // X^{-T} = cof(X)/det(X); gamma = sqrt(||X^{-1}||_F / ||X||_F)
#pragma unroll
        for (int it = 0; it < 6; ++it) {
            float C0 = X[4] * X[8] - X[5] * X[7];
            float C1 = X[5] * X[6] - X[3] * X[8];
            float C2 = X[3] * X[7] - X[4] * X[6];
            float C3 = X[2] * X[7] - X[1] * X[8];
            float C4 = X[0] * X[8] - X[2] * X[6];
            float C5 = X[1] * X[6] - X[0] * X[7];
            float C6 = X[1] * X[5] - X[2] * X[4];
            float C7 = X[2] * X[3] - X[0] * X[5];
            float C8 = X[0] * X[4] - X[1] * X[3];
            const float det = X[0] * C0 + X[1] * C1 + X[2] * C2;
            const float inv_det = 1.f / det;
            const float nx = X[0]*X[0]+X[1]*X[1]+X[2]*X[2]+X[3]*X[3]+X[4]*X[4]
                           + X[5]*X[5]+X[6]*X[6]+X[7]*X[7]+X[8]*X[8];
            const float nc = C0*C0+C1*C1+C2*C2+C3*C3+C4*C4+C5*C5+C6*C6+C7*C7+C8*C8;
            const float gam = sqrtf(sqrtf(nc) / (fabsf(det) * sqrtf(nx)));
            const float ig  = 0.5f * inv_det / gam;
            const float hg  = 0.5f * gam;
            X[0] = hg * X[0] + ig * C0;  X[1] = hg * X[1] + ig * C1;
            X[2] = hg * X[2] + ig * C2;  X[3] = hg * X[3] + ig * C3;
            X[4] = hg * X[4] + ig * C4;  X[5] = hg * X[5] + ig * C5;
            X[6] = hg * X[6] + ig * C6;  X[7] = hg * X[7] + ig * C7;
            X[8] = hg * X[8] + ig * C8;
        }

#pragma unroll
        for (int j = 0; j < 9; ++j) out[(size_t)g * 9 + j] = X[j];
        logdet[g] = 0.f;
    }
}

extern "C" void kernel_launch(void* const* d_in, const int* in_sizes, int n_in,
                              void* d_out, int out_size, void* d_ws, size_t ws_size,
                              hipStream_t stream) {
    const float* rotation = (const float*)d_in[0];
    const float* feature  = (const float*)d_in[1];
    const float* W1 = (const float*)d_in[2];
    const float* b1 = (const float*)d_in[3];
    const float* W2 = (const float*)d_in[4];
    const float* b2 = (const float*)d_in[5];
    const float* W3 = (const float*)d_in[6];
    const float* b3 = (const float*)d_in[7];

    float* out    = (float*)d_out;                 // [N,9]
    float* logdet = out + (size_t)N_PTS * 9;       // [N]

    const size_t smemBytes =
        (size_t)(M_TILE * SF_STRIDE + 2 * M_TILE * SH_STRIDE) * sizeof(float); // ~167 KB

    dim3 grid(N_PTS / M_TILE);
    fused_rot_mlp_polar<<<grid, TPB, smemBytes, stream>>>(
        rotation, feature, W1, b1, W2, b2, W3, b3, out, logdet);
}